// singleESN_22771916603668
// MI455X (gfx1250) — compile-verified
//
#include <hip/hip_runtime.h>
#include <math.h>

// ===========================================================================
// Leaky-integrator ESN for MI455X (gfx1250):
//   Phase 1: uproj = x @ Win    -> fp32 GEMM on the WMMA pipe (16x16x4 f32)
//   Phase 2: sequential scan    -> persistent grid, 4-way K-split matvec,
//                                  async-to-LDS v staging, grid barrier/step
// uproj is written straight into d_out and overwritten row-by-row with states.
// ===========================================================================

typedef __attribute__((ext_vector_type(2))) float v2f;
typedef __attribute__((ext_vector_type(8))) float v8f;

#define T_STEPS 4096
#define N_IN    4096
#define N_RES   2048
#define ALPHA_F 0.5f

// --------------------------- Phase 1: WMMA GEMM ----------------------------
#define BM 64
#define BN 128
#define BK 32

__global__ __launch_bounds__(256) void proj_gemm_f32_wmma(
    const float* __restrict__ A,   // x   [T_STEPS][N_IN]
    const float* __restrict__ B,   // Win [N_IN][N_RES]
    float* __restrict__ C)         // out [T_STEPS][N_RES]
{
  __shared__ float As[BM][BK + 1];
  __shared__ float Bs[BN][BK + 1];   // transposed: [n][k]

  const int mBase = blockIdx.y * BM;
  const int nBase = blockIdx.x * BN;
  const int tid   = threadIdx.x;
  const int lane  = tid & 31;
  const int wv    = tid >> 5;
  const int wm    = (wv & 3) << 4;
  const int wn    = (wv >> 2) << 6;
  const int l15   = lane & 15;
  const int khalf = (lane >> 4) << 1;

  v8f acc0 = {}, acc1 = {}, acc2 = {}, acc3 = {};

  for (int kBase = 0; kBase < N_IN; kBase += BK) {
    for (int i = tid; i < BM * BK; i += 256) {
      const int m = i >> 5, k = i & 31;
      As[m][k] = A[(size_t)(mBase + m) * N_IN + (kBase + k)];
    }
    for (int i = tid; i < BK * BN; i += 256) {
      const int k = i >> 7, n = i & 127;
      Bs[n][k] = B[(size_t)(kBase + k) * N_RES + (nBase + n)];
    }
    __syncthreads();

#pragma unroll
    for (int kk = 0; kk < BK; kk += 4) {
      const int k0 = kk + khalf;
      v2f aF;
      aF.x = As[wm + l15][k0];
      aF.y = As[wm + l15][k0 + 1];
      v2f b0, b1, b2, b3;
      b0.x = Bs[wn +  0 + l15][k0]; b0.y = Bs[wn +  0 + l15][k0 + 1];
      b1.x = Bs[wn + 16 + l15][k0]; b1.y = Bs[wn + 16 + l15][k0 + 1];
      b2.x = Bs[wn + 32 + l15][k0]; b2.y = Bs[wn + 32 + l15][k0 + 1];
      b3.x = Bs[wn + 48 + l15][k0]; b3.y = Bs[wn + 48 + l15][k0 + 1];

      acc0 = __builtin_amdgcn_wmma_f32_16x16x4_f32(false, aF, false, b0,
                                                   (short)0, acc0, false, false);
      acc1 = __builtin_amdgcn_wmma_f32_16x16x4_f32(false, aF, false, b1,
                                                   (short)0, acc1, false, false);
      acc2 = __builtin_amdgcn_wmma_f32_16x16x4_f32(false, aF, false, b2,
                                                   (short)0, acc2, false, false);
      acc3 = __builtin_amdgcn_wmma_f32_16x16x4_f32(false, aF, false, b3,
                                                   (short)0, acc3, false, false);
    }
    __syncthreads();
  }

  const int mrow = mBase + wm + ((lane >> 4) << 3);
#pragma unroll
  for (int r = 0; r < 8; ++r) {
    float* dst = C + (size_t)(mrow + r) * N_RES + nBase + l15;
    dst[wn +  0] = acc0[r];
    dst[wn + 16] = acc1[r];
    dst[wn + 32] = acc2[r];
    dst[wn + 48] = acc3[r];
  }
}

// ------------------------ W transpose (optional path) ----------------------
__global__ __launch_bounds__(256) void transpose_w(
    const float* __restrict__ W, float* __restrict__ Wt)
{
  __shared__ float tile[32][33];
  const int bx = blockIdx.x * 32, by = blockIdx.y * 32;
  const int tx = threadIdx.x, ty = threadIdx.y;   // 32 x 8
#pragma unroll
  for (int j = 0; j < 32; j += 8)
    tile[ty + j][tx] = W[(size_t)(by + ty + j) * N_RES + bx + tx];
  __syncthreads();
#pragma unroll
  for (int j = 0; j < 32; j += 8)
    Wt[(size_t)(bx + ty + j) * N_RES + by + tx] = tile[tx][ty + j];
}

// --------------------------- Phase 2: sequential scan ----------------------
// 32 blocks x 256 threads.  tid = p*64 + c : thread handles column
// col = blockIdx.x*64 + c, K-partition p in [0,4) covering 512 K values.
// v_{t-1} staged to LDS via gfx1250 async-to-LDS (ASYNCcnt), W streamed
// from L2; partial dots reduced through LDS; grid barrier per step.
#define SCAN_BLOCKS  32
#define SCAN_THREADS 256
#define COLS_PER_BLK 64
#define KSPLIT       4
#define KSEG         (N_RES / KSPLIT)   // 512

__device__ __forceinline__ void async_copy_b128(unsigned lds_addr,
                                                const float* __restrict__ g) {
  const unsigned long long ga = (unsigned long long)(uintptr_t)g;
  asm volatile("global_load_async_to_lds_b128 %0, %1, off"
               :: "v"(lds_addr), "v"(ga) : "memory");
}
__device__ __forceinline__ void wait_asynccnt0() {
  asm volatile("s_wait_asynccnt 0" ::: "memory");
}

template <int USE_WT>
__global__ __launch_bounds__(SCAN_THREADS) void esn_scan(
    const float* __restrict__ Wmat,  // USE_WT ? Wt [col][k] : W [k][col]
    float* __restrict__ out,         // [T_STEPS][N_RES]
    unsigned* __restrict__ bar)
{
  __shared__ float vsh[N_RES];                 // full v_{t-1} (8 KB)
  __shared__ float red[KSPLIT * COLS_PER_BLK]; // partial dots  (1 KB)

  const int tid  = threadIdx.x;
  const int c    = tid & (COLS_PER_BLK - 1);   // 0..63
  const int p    = tid >> 6;                   // 0..3 (uniform per wave)
  const int col  = blockIdx.x * COLS_PER_BLK + c;
  const unsigned lds0 = (unsigned)(uintptr_t)(&vsh[0]);  // LDS byte offset

  // t = 0: v_{-1} = 0  ->  v_0 = alpha * tanh(u_0)
  if (tid < COLS_PER_BLK) {
    const float u = out[col];
    out[col] = ALPHA_F * tanhf(u);
  }

  for (int t = 1; t < T_STEPS; ++t) {
    // ---- device-wide barrier: row t-1 fully written by all blocks ----
    __syncthreads();
    if (tid == 0) {
      __hip_atomic_fetch_add(bar, 1u, __ATOMIC_RELEASE, __HIP_MEMORY_SCOPE_AGENT);
      const unsigned target = (unsigned)t * SCAN_BLOCKS;
      while (__hip_atomic_load(bar, __ATOMIC_ACQUIRE, __HIP_MEMORY_SCOPE_AGENT) < target)
        __builtin_amdgcn_s_sleep(1);
    }
    __syncthreads();

    // ---- async-stage v_{t-1} into LDS (b128 per thread x2) ----
    const float* __restrict__ vrow = out + (size_t)(t - 1) * N_RES;
#pragma unroll
    for (int i = tid; i < N_RES / 4; i += SCAN_THREADS)
      async_copy_b128(lds0 + 16u * (unsigned)i, vrow + 4 * i);
    wait_asynccnt0();
    __syncthreads();

    // ---- partial dot over K segment [p*512, p*512+512) ----
    float a0 = 0.f, a1 = 0.f, a2 = 0.f, a3 = 0.f;
    float a4 = 0.f, a5 = 0.f, a6 = 0.f, a7 = 0.f;
    const int kb = p * KSEG;
    if (USE_WT) {
      // Wt row for this column: contiguous, stream as float4 pairs
      const float4* __restrict__ wr =
          (const float4*)(Wmat + (size_t)col * N_RES + kb);
      for (int k = 0; k < KSEG; k += 8) {
        const float4 w0 = wr[(k >> 2) + 0];
        const float4 w1 = wr[(k >> 2) + 1];
        a0 += vsh[kb + k + 0] * w0.x;
        a1 += vsh[kb + k + 1] * w0.y;
        a2 += vsh[kb + k + 2] * w0.z;
        a3 += vsh[kb + k + 3] * w0.w;
        a4 += vsh[kb + k + 4] * w1.x;
        a5 += vsh[kb + k + 5] * w1.y;
        a6 += vsh[kb + k + 6] * w1.z;
        a7 += vsh[kb + k + 7] * w1.w;
      }
    } else {
      // column read, coalesced across lanes, immediate offsets j*8192B
      const float* __restrict__ wp = Wmat + (size_t)kb * N_RES + col;
      for (int k = 0; k < KSEG; k += 8) {
        a0 += vsh[kb + k + 0] * wp[0 * N_RES];
        a1 += vsh[kb + k + 1] * wp[1 * N_RES];
        a2 += vsh[kb + k + 2] * wp[2 * N_RES];
        a3 += vsh[kb + k + 3] * wp[3 * N_RES];
        a4 += vsh[kb + k + 4] * wp[4 * N_RES];
        a5 += vsh[kb + k + 5] * wp[5 * N_RES];
        a6 += vsh[kb + k + 6] * wp[6 * N_RES];
        a7 += vsh[kb + k + 7] * wp[7 * N_RES];
        wp += 8 * N_RES;
      }
    }
    red[p * COLS_PER_BLK + c] =
        ((a0 + a1) + (a2 + a3)) + ((a4 + a5) + (a6 + a7));
    __syncthreads();

    // ---- reduce 4 partials, leaky tanh update into row t ----
    if (tid < COLS_PER_BLK) {
      const float dot = ((red[0 * COLS_PER_BLK + tid] +
                          red[1 * COLS_PER_BLK + tid]) +
                         (red[2 * COLS_PER_BLK + tid] +
                          red[3 * COLS_PER_BLK + tid]));
      const float u     = out[(size_t)t * N_RES + col];
      const float vprev = vsh[col];
      out[(size_t)t * N_RES + col] =
          (1.0f - ALPHA_F) * vprev + ALPHA_F * tanhf(dot + u);
    }
  }
}

// --------------------------------- launch ----------------------------------
extern "C" void kernel_launch(void* const* d_in, const int* in_sizes, int n_in,
                              void* d_out, int out_size, void* d_ws, size_t ws_size,
                              hipStream_t stream) {
  const float* x   = (const float*)d_in[0];  // [4096,4096]
  const float* Win = (const float*)d_in[1];  // [4096,2048]
  const float* W   = (const float*)d_in[2];  // [2048,2048]
  float* out = (float*)d_out;                // [4096,2048]

  // Phase 1: uproj -> d_out rows
  dim3 grid(N_RES / BN, T_STEPS / BM);       // (16, 64)
  proj_gemm_f32_wmma<<<grid, 256, 0, stream>>>(x, Win, out);

  // Barrier counter lives at d_ws[0]; W^T (optional) at d_ws+256.
  hipMemsetAsync(d_ws, 0, 256, stream);

  const size_t WT_BYTES = (size_t)N_RES * N_RES * sizeof(float);
  unsigned* bar = (unsigned*)d_ws;
  if (ws_size >= 256 + WT_BYTES) {
    float* Wt = (float*)((char*)d_ws + 256);
    transpose_w<<<dim3(N_RES / 32, N_RES / 32), dim3(32, 8), 0, stream>>>(W, Wt);
    esn_scan<1><<<SCAN_BLOCKS, SCAN_THREADS, 0, stream>>>(Wt, out, bar);
  } else {
    esn_scan<0><<<SCAN_BLOCKS, SCAN_THREADS, 0, stream>>>(W, out, bar);
  }
}